// SpatialTemporalAttention_37529424232830
// MI455X (gfx1250) — compile-verified
//
#include <hip/hip_runtime.h>

typedef __attribute__((ext_vector_type(2))) float v2f;
typedef __attribute__((ext_vector_type(8))) float v8f;
typedef __attribute__((ext_vector_type(4))) unsigned v4u;
typedef __attribute__((ext_vector_type(8))) unsigned v8u;

#define B_SZ 4
#define T_LEN 1024
#define D_SZ 256
#define H_SZ 4
#define DK 64
#define KB 32   // key block for flash attention

// ---------------------------------------------------------------------------
// Tensor Data Mover: 2D tile load Global->LDS with hardware row padding of
// 4 DWORDs every 64 DWORDs (so a 64-float tile row lands at LDS stride 68).
// Descriptor per CDNA5 ISA sec 8.3/8.4. Call from ONE wave, uniform args only;
// follow with s_wait_tensorcnt(0) + __syncthreads() before consuming.
// ---------------------------------------------------------------------------
__device__ __forceinline__ void tdm_load_2d(unsigned lds_addr, const void* gptr,
                                            unsigned rowlen, unsigned nrows,
                                            unsigned stride, unsigned tile_w,
                                            unsigned tile_h) {
  unsigned long long ga = (unsigned long long)(uintptr_t)gptr;
  v4u g0 = {1u,                                   // count=1 valid descriptor
            lds_addr,                             // lds_addr[31:0]
            (unsigned)ga,                         // global_addr[31:0]
            (unsigned)((ga >> 32) & 0x1FFFFFFu) | (2u << 30)};  // addr[56:32],type=2
  v8u g1 = {(2u << 16)          // data_size = 4 bytes
                | (1u << 20)    // pad_enable
                | (5u << 22)    // pad_interval: 64 DWORDs
                | (3u << 25),   // pad_amount:   4 DWORDs
            (rowlen & 0xFFFFu) << 16,                       // tensor_dim0 lo16
            (rowlen >> 16) | ((nrows & 0xFFFFu) << 16),     // dim0 hi / dim1 lo
            (nrows >> 16) | ((tile_w & 0xFFFFu) << 16),     // dim1 hi / tile_dim0
            (tile_h & 0xFFFFu),                             // tile_dim1 (tile_dim2=0)
            stride,                                         // tensor_dim0_stride lo32
            0u, 0u};
  asm volatile("tensor_load_to_lds %0, %1" ::"s"(g0), "s"(g1) : "memory");
}

// ---------------------------------------------------------------------------
// Kernel 1: qkv = x @ Wqkv^T + bqkv, scattered into q/k/v as (B,H,T,DK).
// Block = 64x32 output tile (8 waves x one 16x16 WMMA tile); A K-panels
// (64x64) staged in LDS via TDM; B (weights) streamed from global/L2.
// ---------------------------------------------------------------------------
__global__ __launch_bounds__(256) void qkv_kernel(
    const float* __restrict__ x, const float* __restrict__ Wqkv,
    const float* __restrict__ bqkv, float* __restrict__ q,
    float* __restrict__ k, float* __restrict__ v) {
  constexpr int AS = 68;
  __shared__ float As[64 * AS];
  const int tid = threadIdx.x;
  const int lane = tid & 31;
  const int wave = tid >> 5;
  const int half = lane >> 4;
  const int l = lane & 15;

  const int NBLK = (3 * D_SZ) / 32;          // 24 column blocks
  const int mB = blockIdx.x / NBLK;
  const int nB = blockIdx.x % NBLK;
  const int m0 = mB * 64 + (wave >> 1) * 16; // this wave's output rows
  const int n0 = nB * 32 + (wave & 1) * 16;  // this wave's output cols

  v8f acc = {};
  const float* brow = Wqkv + (size_t)(n0 + l) * D_SZ;
  const float* ar = &As[((wave >> 1) * 16 + l) * AS];
  for (int kp = 0; kp < D_SZ; kp += 64) {
    if (wave == 0) {
      tdm_load_2d((unsigned)(uintptr_t)&As[0],
                  x + (size_t)mB * 64 * D_SZ + kp,
                  D_SZ, B_SZ * T_LEN, D_SZ, 64, 64);
      __builtin_amdgcn_s_wait_tensorcnt(0);
    }
    __syncthreads();
    for (int k0 = 0; k0 < 64; k0 += 4) {
      const int kk = k0 + half * 2;
      v2f a = {ar[kk], ar[kk + 1]};
      v2f b = {brow[kp + kk], brow[kp + kk + 1]};
      acc = __builtin_amdgcn_wmma_f32_16x16x4_f32(false, a, false, b,
                                                  (short)0, acc, false, false);
    }
    __syncthreads();
  }

  const int n = n0 + l;
  const float bias = bqkv[n];
  const int which = n >> 8;                 // 0:q 1:k 2:v (uniform per tile)
  const int h = (n & 255) >> 6;             // head (uniform per tile)
  const int dk = n & 63;
  float* dst = (which == 0) ? q : (which == 1) ? k : v;
#pragma unroll
  for (int r = 0; r < 8; ++r) {
    const int m = m0 + half * 8 + r;
    const int b_ = m >> 10;
    const int t = m & (T_LEN - 1);
    dst[(((size_t)b_ * H_SZ + h) * T_LEN + t) * DK + dk] = acc[r] + bias;
  }
}

// ---------------------------------------------------------------------------
// Kernel 2: flash attention per (b,h). Grid (T/64, B*H), 256 threads.
// Q (64x64) and each K/V block (32x64) staged via TDM with hardware padding.
// ---------------------------------------------------------------------------
__global__ __launch_bounds__(256) void attn_kernel(
    const float* __restrict__ q, const float* __restrict__ k,
    const float* __restrict__ v, float* __restrict__ ctx) {
  constexpr int QS = 68;   // TDM-padded row stride for Qs/Ks/Vs
  constexpr int SS = 36;   // row stride for score/P tile
  __shared__ float Qs[64 * QS];
  __shared__ float Ks[KB * QS];
  __shared__ float Vs[KB * QS];
  __shared__ float Ps[64 * SS];
  __shared__ float row_m[64], row_l[64], row_scale[64];

  const int qt = blockIdx.x;
  const int bh = blockIdx.y;
  const int tid = threadIdx.x;
  const int lane = tid & 31;
  const int wave = tid >> 5;
  const int half = lane >> 4;
  const int l = lane & 15;
  const int qr = wave >> 1;                  // query-row group (0..3)
  const int cg = wave & 1;                   // column-group select

  const float* qbase = q + ((size_t)bh * T_LEN + qt * 64) * DK;
  const float* kbase = k + (size_t)bh * T_LEN * DK;
  const float* vbase = v + (size_t)bh * T_LEN * DK;

  if (wave == 0) {
    tdm_load_2d((unsigned)(uintptr_t)&Qs[0], qbase, DK, T_LEN, DK, 64, 64);
    __builtin_amdgcn_s_wait_tensorcnt(0);
  }
  if (tid < 64) { row_m[tid] = -1e30f; row_l[tid] = 0.f; }
  v8f accA = {};   // output cols cg*32 + 0..15
  v8f accB = {};   // output cols cg*32 + 16..31
  __syncthreads();

  for (int j = 0; j < T_LEN; j += KB) {
    if (wave == 0) {
      tdm_load_2d((unsigned)(uintptr_t)&Ks[0], kbase + (size_t)j * DK,
                  DK, T_LEN, DK, 64, KB);
      tdm_load_2d((unsigned)(uintptr_t)&Vs[0], vbase + (size_t)j * DK,
                  DK, T_LEN, DK, 64, KB);
      __builtin_amdgcn_s_wait_tensorcnt(0);
    }
    __syncthreads();

    // S = Q K^T / sqrt(DK); each wave computes one 16x16 S tile
    {
      const int m0 = qr * 16;
      const int c0 = cg * 16;
      v8f s = {};
      const float* ar = &Qs[(m0 + l) * QS];
      const float* br = &Ks[(c0 + l) * QS];      // B[kk][col] = K[col][kk]
      for (int k0 = 0; k0 < DK; k0 += 4) {
        const int kk = k0 + half * 2;
        v2f a = {ar[kk], ar[kk + 1]};
        v2f b = {br[kk], br[kk + 1]};
        s = __builtin_amdgcn_wmma_f32_16x16x4_f32(false, a, false, b,
                                                  (short)0, s, false, false);
      }
#pragma unroll
      for (int r = 0; r < 8; ++r)
        Ps[(m0 + half * 8 + r) * SS + c0 + l] = s[r] * 0.125f;  // 1/sqrt(64)
    }
    __syncthreads();

    // online softmax: thread r owns row r
    if (tid < 64) {
      float* srow = &Ps[tid * SS];
      const float mold = row_m[tid];
      float mj = mold;
      for (int c = 0; c < KB; ++c) mj = fmaxf(mj, srow[c]);
      const float scl = __expf(mold - mj);
      float ls = 0.f;
      for (int c = 0; c < KB; ++c) {
        const float e = __expf(srow[c] - mj);
        srow[c] = e;
        ls += e;
      }
      row_m[tid] = mj;
      row_l[tid] = row_l[tid] * scl + ls;
      row_scale[tid] = scl;
    }
    __syncthreads();

    // O = O*scale + P @ V  (two 16x16 output tiles per wave)
    {
      const int m0 = qr * 16;
#pragma unroll
      for (int r = 0; r < 8; ++r) {
        const float scl = row_scale[m0 + half * 8 + r];
        accA[r] *= scl;
        accB[r] *= scl;
      }
      const float* ar = &Ps[(m0 + l) * SS];
      const int c0 = cg * 32;
      for (int k0 = 0; k0 < KB; k0 += 4) {
        const int kk = k0 + half * 2;
        v2f a = {ar[kk], ar[kk + 1]};
        v2f b0 = {Vs[kk * QS + c0 + l], Vs[(kk + 1) * QS + c0 + l]};
        accA = __builtin_amdgcn_wmma_f32_16x16x4_f32(false, a, false, b0,
                                                     (short)0, accA, false, false);
        v2f b1 = {Vs[kk * QS + c0 + 16 + l], Vs[(kk + 1) * QS + c0 + 16 + l]};
        accB = __builtin_amdgcn_wmma_f32_16x16x4_f32(false, a, false, b1,
                                                     (short)0, accB, false, false);
      }
    }
    __syncthreads();
  }

  // normalize + write ctx (B,T,D) with col = h*64 + d
  const int b_ = bh >> 2, h = bh & 3;
  const int m0 = qr * 16;
  const int c0 = cg * 32;
#pragma unroll
  for (int r = 0; r < 8; ++r) {
    const int row = m0 + half * 8 + r;
    const int t = qt * 64 + row;
    const float inv = 1.f / row_l[row];
    float* orow = ctx + ((size_t)b_ * T_LEN + t) * D_SZ + h * DK;
    orow[c0 + l] = accA[r] * inv;
    orow[c0 + 16 + l] = accB[r] * inv;
  }
}

// ---------------------------------------------------------------------------
// Kernel 3: out = ctx @ Wproj^T + bproj.  Same TDM-staged GEMM blocking.
// ---------------------------------------------------------------------------
__global__ __launch_bounds__(256) void proj_kernel(
    const float* __restrict__ ctx, const float* __restrict__ Wproj,
    const float* __restrict__ bproj, float* __restrict__ out) {
  constexpr int AS = 68;
  __shared__ float As[64 * AS];
  const int tid = threadIdx.x;
  const int lane = tid & 31;
  const int wave = tid >> 5;
  const int half = lane >> 4;
  const int l = lane & 15;

  const int NBLK = D_SZ / 32;                // 8 column blocks
  const int mB = blockIdx.x / NBLK;
  const int nB = blockIdx.x % NBLK;
  const int m0 = mB * 64 + (wave >> 1) * 16;
  const int n0 = nB * 32 + (wave & 1) * 16;

  v8f acc = {};
  const float* brow = Wproj + (size_t)(n0 + l) * D_SZ;
  const float* ar = &As[((wave >> 1) * 16 + l) * AS];
  for (int kp = 0; kp < D_SZ; kp += 64) {
    if (wave == 0) {
      tdm_load_2d((unsigned)(uintptr_t)&As[0],
                  ctx + (size_t)mB * 64 * D_SZ + kp,
                  D_SZ, B_SZ * T_LEN, D_SZ, 64, 64);
      __builtin_amdgcn_s_wait_tensorcnt(0);
    }
    __syncthreads();
    for (int k0 = 0; k0 < 64; k0 += 4) {
      const int kk = k0 + half * 2;
      v2f a = {ar[kk], ar[kk + 1]};
      v2f b = {brow[kp + kk], brow[kp + kk + 1]};
      acc = __builtin_amdgcn_wmma_f32_16x16x4_f32(false, a, false, b,
                                                  (short)0, acc, false, false);
    }
    __syncthreads();
  }
  const int n = n0 + l;
  const float bias = bproj[n];
#pragma unroll
  for (int r = 0; r < 8; ++r) {
    const int m = m0 + half * 8 + r;
    out[(size_t)m * D_SZ + n] = acc[r] + bias;
  }
}

// ---------------------------------------------------------------------------
// Kernel 4: spatial (feature-pairwise) attention; rank-1 scores -> pure VALU.
// ---------------------------------------------------------------------------
__global__ __launch_bounds__(256) void spatial_kernel(
    const float* __restrict__ x, float* __restrict__ out) {
  __shared__ float xv[D_SZ];
  const int bt = blockIdx.x;
  const int i = threadIdx.x;
  xv[i] = x[(size_t)bt * D_SZ + i];
  __syncthreads();
  const float xi = xv[i] * 0.0625f;          // 1/sqrt(256)
  float mx = -1e30f;
  for (int j = 0; j < D_SZ; ++j) mx = fmaxf(mx, xi * xv[j]);
  float s = 0.f, acc = 0.f;
  for (int j = 0; j < D_SZ; ++j) {
    const float e = __expf(xi * xv[j] - mx);
    s += e;
    acc += e * xv[j];
  }
  out[(size_t)bt * D_SZ + i] += acc / s;
}

// ---------------------------------------------------------------------------
extern "C" void kernel_launch(void* const* d_in, const int* in_sizes, int n_in,
                              void* d_out, int out_size, void* d_ws, size_t ws_size,
                              hipStream_t stream) {
  (void)in_sizes; (void)n_in; (void)out_size; (void)ws_size;
  const float* x     = (const float*)d_in[0];
  const float* Wqkv  = (const float*)d_in[1];
  const float* bqkv  = (const float*)d_in[2];
  const float* Wproj = (const float*)d_in[3];
  const float* bproj = (const float*)d_in[4];
  float* out = (float*)d_out;

  const size_t QKV_ELEMS = (size_t)B_SZ * H_SZ * T_LEN * DK;   // 1M floats each
  float* q   = (float*)d_ws;
  float* k   = q + QKV_ELEMS;
  float* v   = k + QKV_ELEMS;
  float* ctx = v + QKV_ELEMS;

  // 1) QKV projection: (4096/64) x (768/32) blocks
  qkv_kernel<<<64 * 24, 256, 0, stream>>>(x, Wqkv, bqkv, q, k, v);
  // 2) temporal flash attention: grid (T/64, B*H)
  attn_kernel<<<dim3(T_LEN / 64, B_SZ * H_SZ), 256, 0, stream>>>(q, k, v, ctx);
  // 3) output projection -> out (y_t): (4096/64) x (256/32) blocks
  proj_kernel<<<64 * 8, 256, 0, stream>>>(ctx, Wproj, bproj, out);
  // 4) spatial attention adds y_f into out
  spatial_kernel<<<B_SZ * T_LEN, 256, 0, stream>>>(x, out);
}